// BiLinearInterpolation_3453153706613
// MI455X (gfx1250) — compile-verified
//
#include <hip/hip_runtime.h>

typedef __attribute__((ext_vector_type(2))) float v2f;
typedef __attribute__((ext_vector_type(8))) float v8f;

#define IN_H  512
#define IN_W  512
#define OUT_H 256
#define OUT_W 256
#define CH    16

__global__ __launch_bounds__(256)
void stn_bilinear_wmma_kernel(const float* __restrict__ X,
                              const float* __restrict__ theta,
                              float* __restrict__ out)
{
    const int b    = blockIdx.x >> 8;     // 256 blocks per image
    const int oy   = blockIdx.x & 255;    // one output row per block
    const int ox   = threadIdx.x;         // 256 columns
    const int lane = threadIdx.x & 31;
    const int ox_base = ox & ~31;         // first column of this wave

    // theta row: uniform per block -> scalarized by compiler
    const float t0 = theta[b * 6 + 0];
    const float t1 = theta[b * 6 + 1];
    const float t2 = theta[b * 6 + 2];
    const float t3 = theta[b * 6 + 3];
    const float t4 = theta[b * 6 + 4];
    const float t5 = theta[b * 6 + 5];

    const float step = 2.0f / 255.0f;           // linspace(-1,1,256) step
    const float gy   = -1.0f + (float)oy * step;

    // ---------------- affine transform via v_wmma_f32_16x16x4_f32 ----------------
    // D = A(16x4) @ B(4x16):
    //   A row0 = (t0,t1,t2,0), A row1 = (t3,t4,t5,0), rows 2..15 = 0
    //   B row0 = gx(col), row1 = gy, row2 = 1, row3 = 0
    // A layout (ISA 16x4 f32): VGPR0 = {K0 lanes0-15 | K2 lanes16-31}, VGPR1 = {K1 | K3}
    v2f a;
    a.x = (lane == 0) ? t0 : (lane == 1) ? t3 :
          (lane == 16) ? t2 : (lane == 17) ? t5 : 0.0f;
    a.y = (lane == 0) ? t1 : (lane == 1) ? t4 : 0.0f;

    const int   col   = lane & 15;
    const float gx_lo = -1.0f + (float)(ox_base + col)      * step;
    const float gx_hi = -1.0f + (float)(ox_base + 16 + col) * step;

    // B layout mirrors A: VGPR0 = {K0 | K2}, VGPR1 = {K1 | K3}
    v2f b0, b1;
    b0.x = (lane < 16) ? gx_lo : 1.0f;   // K0 = gx, K2 = 1
    b0.y = (lane < 16) ? gy    : 0.0f;   // K1 = gy, K3 = 0
    b1.x = (lane < 16) ? gx_hi : 1.0f;
    b1.y = (lane < 16) ? gy    : 0.0f;

    v8f cz = {0.0f, 0.0f, 0.0f, 0.0f, 0.0f, 0.0f, 0.0f, 0.0f};
    // 8 args: (neg_a, A, neg_b, B, c_mod, C, reuse_a, reuse_b)
    v8f d0 = __builtin_amdgcn_wmma_f32_16x16x4_f32(false, a, false, b0, (short)0, cz, false, false);
    v8f d1 = __builtin_amdgcn_wmma_f32_16x16x4_f32(false, a, false, b1, (short)0, cz, false, false);

    // C/D layout: VGPR0 = row M=0 (lanes0-15), VGPR1 = row M=1 -> lane n<16 holds
    // (sgx,sgy) of pixel ox_base+n in d0[0],d0[1]; d1 covers ox_base+16..31.
    float hx = __shfl(d1[0], lane - 16, 32);
    float hy = __shfl(d1[1], lane - 16, 32);
    float sgx = (lane < 16) ? d0[0] : hx;
    float sgy = (lane < 16) ? d0[1] : hy;
    // -----------------------------------------------------------------------------

    float xf = 0.5f * (sgx + 1.0f) * (float)IN_W;
    float yf = 0.5f * (sgy + 1.0f) * (float)IN_H;

    int x0 = (int)xf;          // trunc toward zero, matches astype(int32)
    int x1 = x0 + 1;
    int y0 = (int)yf;
    int y1 = y0 + 1;
    x0 = min(max(x0, 0), IN_W - 1);
    x1 = min(max(x1, 0), IN_W - 1);
    y0 = min(max(y0, 0), IN_H - 1);
    y1 = min(max(y1, 0), IN_H - 1);

    // weights from CLAMPED integer coords (matches reference exactly)
    const float x0f = (float)x0, x1f = (float)x1;
    const float y0f = (float)y0, y1f = (float)y1;
    const float wa = (x1f - xf) * (y1f - yf);
    const float wb = (x1f - xf) * (yf - y0f);
    const float wc = (xf - x0f) * (y1f - yf);
    const float wd = (xf - x0f) * (yf - y0f);

    const size_t img = (size_t)b * (size_t)(IN_H * IN_W * CH);
    const float4* pa = (const float4*)(X + img + ((size_t)y0 * IN_W + (size_t)x0) * CH);
    const float4* pb = (const float4*)(X + img + ((size_t)y1 * IN_W + (size_t)x0) * CH);
    const float4* pc = (const float4*)(X + img + ((size_t)y0 * IN_W + (size_t)x1) * CH);
    const float4* pd = (const float4*)(X + img + ((size_t)y1 * IN_W + (size_t)x1) * CH);

    float4* o = (float4*)(out + ((size_t)blockIdx.x * OUT_W + (size_t)ox) * CH);

#pragma unroll
    for (int q = 0; q < 4; ++q) {
        const float4 A4 = pa[q];
        const float4 B4 = pb[q];
        const float4 C4 = pc[q];
        const float4 D4 = pd[q];
        float4 r;
        r.x = fmaf(wd, D4.x, fmaf(wc, C4.x, fmaf(wb, B4.x, wa * A4.x)));
        r.y = fmaf(wd, D4.y, fmaf(wc, C4.y, fmaf(wb, B4.y, wa * A4.y)));
        r.z = fmaf(wd, D4.z, fmaf(wc, C4.z, fmaf(wb, B4.z, wa * A4.z)));
        r.w = fmaf(wd, D4.w, fmaf(wc, C4.w, fmaf(wb, B4.w, wa * A4.w)));
        o[q] = r;
    }
}

extern "C" void kernel_launch(void* const* d_in, const int* in_sizes, int n_in,
                              void* d_out, int out_size, void* d_ws, size_t ws_size,
                              hipStream_t stream) {
    const float* X     = (const float*)d_in[0];   // [16, 512, 512, 16] f32
    const float* theta = (const float*)d_in[1];   // [16, 6] f32
    float* out         = (float*)d_out;           // [16, 256, 256, 16] f32

    (void)in_sizes; (void)n_in; (void)out_size; (void)d_ws; (void)ws_size;

    dim3 grid(16 * OUT_H);   // one block per (image, output row)
    dim3 block(OUT_W);       // 256 threads = 8 waves, all full (EXEC all-1s for WMMA)
    stn_bilinear_wmma_kernel<<<grid, block, 0, stream>>>(X, theta, out);
}